// EuclideanGMM_1580547969984
// MI455X (gfx1250) — compile-verified
//
#include <hip/hip_runtime.h>
#include <stdint.h>

typedef __attribute__((ext_vector_type(2))) float v2f;
typedef __attribute__((ext_vector_type(8))) float v8f;

// exp(-0.5*x) = exp2(x * -0.5*log2(e))
#define NEG_HALF_LOG2E (-0.72134752044448170368f)

__global__ __launch_bounds__(256) void gmm_l2_kernel(
    const float* __restrict__ w1, const float* __restrict__ mu1, const float* __restrict__ sg1,
    const float* __restrict__ w2, const float* __restrict__ mu2, const float* __restrict__ sg2,
    float* __restrict__ partial)
{
    // Double-buffered a-side (set1 in A, set2 in B), filled by async global->LDS DMA.
    __shared__ __align__(16) float s_muA[128 * 32];
    __shared__ __align__(16) float s_sgA[128 * 32];
    __shared__ __align__(16) float s_muB[128 * 32];
    __shared__ __align__(16) float s_sgB[128 * 32];
    __shared__ float s_w1[128];
    __shared__ float s_w2[128];
    __shared__ float s_part[8];

    const int b    = blockIdx.x;
    const int t    = threadIdx.x;
    const int n    = t & 127;     // column owned by this thread
    const int base = t >> 7;      // row parity (0 or 1)
    const int lane = t & 31;
    const int wave = t >> 5;

    const size_t bK = (size_t)b * 4096;   // 128*32 elements per batch per array

    // ---- async global->LDS staging: 4096 floats = 1024 B128 ops per array,
    //      256 threads -> 4 wave-wide async instructions per array per thread ----
    auto asyncCopy = [&](const float* g, float* s) {
        uint64_t ga = (uint64_t)(uintptr_t)g + (uint64_t)t * 16u;
        uint32_t la = (uint32_t)(uintptr_t)(void*)s + (uint32_t)t * 16u;
        #pragma unroll
        for (int q = 0; q < 4; ++q) {
            asm volatile("global_load_async_to_lds_b128 %0, %1, off"
                         :: "v"(la), "v"(ga) : "memory");
            ga += 4096u;   // 256 threads * 16B
            la += 4096u;
        }
    };
    asyncCopy(mu1 + bK, s_muA);   // ASYNCcnt ops 1..4
    asyncCopy(sg1 + bK, s_sgA);   // 5..8
    asyncCopy(mu2 + bK, s_muB);   // 9..12   (prefetched under phases 1-2)
    asyncCopy(sg2 + bK, s_sgB);   // 13..16

    // ---- stage weights (plain DS stores, covered by the barrier) ----
    if (t < 128) {
        s_w1[t] = w1[(size_t)b * 128 + t];
        s_w2[t] = w2[(size_t)b * 128 + t];
    }
    const float wb1 = w1[(size_t)b * 128 + n];
    const float wb2 = w2[(size_t)b * 128 + n];

    // ---- b-side row cached in registers (this thread's column n) ----
    float4 bm[8], bs[8];
    auto loadB = [&](const float* mu, const float* sg) {
        const float4* gm = (const float4*)(mu + bK + (size_t)n * 32);
        const float4* gs = (const float4*)(sg + bK + (size_t)n * 32);
        #pragma unroll
        for (int q = 0; q < 8; ++q) { bm[q] = gm[q]; bs[q] = gs[q]; }
    };

    // ---- WMMA accumulator: all-ones B => one wmma folds 64 values into C ----
    v8f c = {};
    v2f bones; bones.x = 1.0f; bones.y = 1.0f;

    auto doPhase = [&](float wbEff, const float* s_wa,
                       const float* smu, const float* ssg) {
        for (int ii = 0; ii < 64; ii += 2) {
            float v0 = 0.f, v1 = 0.f;
            #pragma unroll
            for (int h = 0; h < 2; ++h) {
                const int m = (base + 2 * (ii + h)) & 127;
                const float4* am4 = (const float4*)(smu + m * 32);  // broadcast reads
                const float4* as4 = (const float4*)(ssg + m * 32);
                float a0 = 0.f, a1 = 0.f, a2 = 0.f, a3 = 0.f;
                #pragma unroll
                for (int q = 0; q < 8; ++q) {
                    const float4 am = am4[q];
                    const float4 as = as4[q];
                    const float dx = am.x - bm[q].x, dy = am.y - bm[q].y;
                    const float dz = am.z - bm[q].z, dw = am.w - bm[q].w;
                    const float sx = as.x + bs[q].x, sy = as.y + bs[q].y;
                    const float sz = as.z + bs[q].z, sw = as.w + bs[q].w;
                    a0 += dx * dx * __builtin_amdgcn_rcpf(sx);
                    a1 += dy * dy * __builtin_amdgcn_rcpf(sy);
                    a2 += dz * dz * __builtin_amdgcn_rcpf(sz);
                    a3 += dw * dw * __builtin_amdgcn_rcpf(sw);
                }
                const float acc = (a0 + a1) + (a2 + a3);
                const float val = wbEff * s_wa[m] *
                                  __builtin_amdgcn_exp2f(acc * NEG_HALF_LOG2E);
                if (h == 0) v0 = val; else v1 = val;
            }
            v2f A; A.x = v0; A.y = v1;
            // D = A x Ones + C : row-sums of A accumulate into C (exact; /16 at end)
            c = __builtin_amdgcn_wmma_f32_16x16x4_f32(
                    false, A, false, bones, (short)0, c, false, false);
        }
    };

    // ---- Phase 1: d11 (a=set1 bufA, b=set1 regs, +w1[m]*w1[n]) ----
    loadB(mu1, sg1);
    asm volatile("s_wait_asynccnt 0x8" ::: "memory");  // buffer A landed (in-order)
    __syncthreads();
    doPhase(wb1, s_w1, s_muA, s_sgA);

    // ---- Phase 2: d12 (a=set1 bufA, b=set2 regs, -2*w1[m]*w2[n]) ----
    loadB(mu2, sg2);
    doPhase(-2.0f * wb2, s_w1, s_muA, s_sgA);

    // ---- Phase 3: d22 (a=set2 bufB prefetched async, b=set2 regs, +w2[m]*w2[n]) ----
    asm volatile("s_wait_asynccnt 0x0" ::: "memory");  // buffer B landed
    __syncthreads();                                   // all waves' DMA visible
    doPhase(wb2, s_w2, s_muB, s_sgB);

    // ---- reduce WMMA accumulator: wave -> workgroup -> d_ws[b] ----
    float s = ((c[0] + c[1]) + (c[2] + c[3])) + ((c[4] + c[5]) + (c[6] + c[7]));
    #pragma unroll
    for (int off = 16; off > 0; off >>= 1)
        s += __shfl_xor(s, off, 32);
    if (lane == 0) s_part[wave] = s;
    __syncthreads();
    if (t == 0) {
        float tot = 0.f;
        #pragma unroll
        for (int i = 0; i < 8; ++i) tot += s_part[i];
        partial[b] = tot;
    }
}

__global__ __launch_bounds__(256) void gmm_reduce_kernel(const float* __restrict__ partial,
                                                         float* __restrict__ out)
{
    __shared__ float sp[8];
    const int t = threadIdx.x;
    float v = partial[t];
    #pragma unroll
    for (int off = 16; off > 0; off >>= 1)
        v += __shfl_xor(v, off, 32);
    if ((t & 31) == 0) sp[t >> 5] = v;
    __syncthreads();
    if (t == 0) {
        float tot = 0.f;
        #pragma unroll
        for (int i = 0; i < 8; ++i) tot += sp[i];
        // /16 for WMMA row-sum replication, /256 for mean over batches
        out[0] = tot * (1.0f / 4096.0f);
    }
}

extern "C" void kernel_launch(void* const* d_in, const int* in_sizes, int n_in,
                              void* d_out, int out_size, void* d_ws, size_t ws_size,
                              hipStream_t stream) {
    const float* w1  = (const float*)d_in[0];
    const float* mu1 = (const float*)d_in[1];
    const float* sg1 = (const float*)d_in[2];
    const float* w2  = (const float*)d_in[3];
    const float* mu2 = (const float*)d_in[4];
    const float* sg2 = (const float*)d_in[5];
    float* partial = (float*)d_ws;   // 256 floats of scratch

    gmm_l2_kernel<<<256, 256, 0, stream>>>(w1, mu1, sg1, w2, mu2, sg2, partial);
    gmm_reduce_kernel<<<1, 256, 0, stream>>>(partial, (float*)d_out);
}